// GATConv_38405597561036
// MI455X (gfx1250) — compile-verified
//
#include <hip/hip_runtime.h>
#include <hip/hip_bf16.h>
#include <math.h>

typedef __attribute__((ext_vector_type(2))) float v2f;
typedef __attribute__((ext_vector_type(8))) float v8f;

#define IN_F   256
#define HEADS  4
#define OUT_F  64
#define LDS_STRIDE 260   // 256 + 4 pad -> conflict-free column reads

// ---------------------------------------------------------------------------
// Kernel 1: feat = emb @ W  via V_WMMA_F32_16X16X4_F32.
// Block = 128 threads (4 waves). Block handles a 16-row strip of emb.
// Wave w handles output cols [w*64, w*64+64) as four 16x16 tiles.
// A strip (16x256) staged in LDS with padded stride.
// ---------------------------------------------------------------------------
__global__ __launch_bounds__(128)
void gat_gemm_wmma(const float* __restrict__ emb,
                   const float* __restrict__ W,
                   float* __restrict__ feat) {
    __shared__ float As[16 * LDS_STRIDE];

    const int tid  = threadIdx.x;
    const int m0   = blockIdx.x * 16;

    // Cooperative load of 16x256 f32 (4096 floats = 1024 float4) into LDS.
    const float4* Asrc = (const float4*)(emb + (size_t)m0 * IN_F);
    #pragma unroll
    for (int j = 0; j < 8; ++j) {
        int idx4 = tid + j * 128;            // 0..1023
        int r    = idx4 >> 6;                // row (64 float4 per row)
        int cc   = (idx4 & 63) << 2;         // col in floats
        float4 v = Asrc[idx4];
        *(float4*)&As[r * LDS_STRIDE + cc] = v;
    }
    __syncthreads();

    const int lane = tid & 31;
    const int wave = tid >> 5;
    const int nrow = lane & 15;   // N index for B/C, M index for A frag
    const int hi   = lane >> 4;   // half-wave selector

    v8f acc0 = {}, acc1 = {}, acc2 = {}, acc3 = {};

    const int n_base = wave * 64;
    for (int k0 = 0; k0 < IN_F; k0 += 4) {
        const int kk = k0 + 2 * hi;          // K row this half-wave holds
        // A fragment: lane holds A[nrow][kk], A[nrow][kk+1]
        v2f a;
        a.x = As[nrow * LDS_STRIDE + kk];
        a.y = As[nrow * LDS_STRIDE + kk + 1];

        // B fragments for four 16-col tiles: B[kk][n], B[kk+1][n]
        const float* Wp = W + (size_t)kk * IN_F + n_base + nrow;
        v2f b0, b1, b2, b3;
        b0.x = Wp[0];          b0.y = Wp[IN_F];
        b1.x = Wp[16];         b1.y = Wp[IN_F + 16];
        b2.x = Wp[32];         b2.y = Wp[IN_F + 32];
        b3.x = Wp[48];         b3.y = Wp[IN_F + 48];

        acc0 = __builtin_amdgcn_wmma_f32_16x16x4_f32(false, a, false, b0, (short)0, acc0, false, false);
        acc1 = __builtin_amdgcn_wmma_f32_16x16x4_f32(false, a, false, b1, (short)0, acc1, false, false);
        acc2 = __builtin_amdgcn_wmma_f32_16x16x4_f32(false, a, false, b2, (short)0, acc2, false, false);
        acc3 = __builtin_amdgcn_wmma_f32_16x16x4_f32(false, a, false, b3, (short)0, acc3, false, false);
    }

    // Store C/D: VGPR i -> row m0 + hi*8 + i, col n_base + t*16 + nrow.
    float* out_base = feat + (size_t)(m0 + hi * 8) * IN_F + n_base + nrow;
    #pragma unroll
    for (int i = 0; i < 8; ++i) {
        out_base[(size_t)i * IN_F +  0] = acc0[i];
        out_base[(size_t)i * IN_F + 16] = acc1[i];
        out_base[(size_t)i * IN_F + 32] = acc2[i];
        out_base[(size_t)i * IN_F + 48] = acc3[i];
    }
}

// ---------------------------------------------------------------------------
// Kernel 2: el[n,h] = sum_d feat[n,h,d]*attn_l[h,d]; er likewise.
// One wave per node; lane l covers 8 contiguous feats; head = l>>3.
// Width-8 shuffle reduction within each head group.
// ---------------------------------------------------------------------------
__global__ __launch_bounds__(256)
void gat_elr(const float* __restrict__ feat,
             const float* __restrict__ attn_l,
             const float* __restrict__ attn_r,
             float* __restrict__ el, float* __restrict__ er, int N) {
    const int lane = threadIdx.x & 31;
    const int wave = threadIdx.x >> 5;
    const int n    = blockIdx.x * 8 + wave;
    if (n >= N) return;

    const int head = lane >> 3;
    const int off  = (lane & 7) * 8;                 // offset within head
    const float* f = feat + (size_t)n * IN_F + lane * 8;
    const float* al = attn_l + head * OUT_F + off;
    const float* ar = attn_r + head * OUT_F + off;

    float sl = 0.f, sr = 0.f;
    #pragma unroll
    for (int j = 0; j < 8; ++j) {
        float v = f[j];
        sl += v * al[j];
        sr += v * ar[j];
    }
    #pragma unroll
    for (int d = 4; d > 0; d >>= 1) {
        sl += __shfl_xor(sl, d, 8);
        sr += __shfl_xor(sr, d, 8);
    }
    if ((lane & 7) == 0) {
        el[n * HEADS + head] = sl;
        er[n * HEADS + head] = sr;
    }
}

// ---------------------------------------------------------------------------
// Kernel 3: init per-(node,head) softmax state.
// ---------------------------------------------------------------------------
__global__ void gat_init_nh(float* __restrict__ m, float* __restrict__ denom, int total) {
    int i = blockIdx.x * blockDim.x + threadIdx.x;
    if (i < total) {
        m[i] = -INFINITY;
        denom[i] = 0.f;
    }
}

// ---------------------------------------------------------------------------
// Kernel 4: edge logits + leaky relu + segment max (native f32 atomic max).
// ---------------------------------------------------------------------------
__global__ void gat_logits(const float* __restrict__ el, const float* __restrict__ er,
                           const int* __restrict__ src, const int* __restrict__ dst,
                           float* __restrict__ ebuf, float* __restrict__ m, int E4) {
    int i = blockIdx.x * blockDim.x + threadIdx.x;
    if (i >= E4) return;
    int e = i >> 2, h = i & 3;
    int s = src[e], d = dst[e];
    float v = el[s * HEADS + h] + er[d * HEADS + h];
    v = (v > 0.f) ? v : 0.2f * v;                    // leaky_relu(0.2)
    ebuf[i] = v;
    __hip_atomic_fetch_max(&m[d * HEADS + h], v,
                           __ATOMIC_RELAXED, __HIP_MEMORY_SCOPE_AGENT);
}

// ---------------------------------------------------------------------------
// Kernel 5: ex = exp(e - m[dst]); denom[dst] += ex (native f32 atomic add).
// ---------------------------------------------------------------------------
__global__ void gat_expsum(const int* __restrict__ dst,
                           float* __restrict__ ebuf,
                           const float* __restrict__ m,
                           float* __restrict__ denom, int E4) {
    int i = blockIdx.x * blockDim.x + threadIdx.x;
    if (i >= E4) return;
    int e = i >> 2, h = i & 3;
    int d = dst[e];
    float ex = __expf(ebuf[i] - m[d * HEADS + h]);
    ebuf[i] = ex;
    __hip_atomic_fetch_add(&denom[d * HEADS + h], ex,
                           __ATOMIC_RELAXED, __HIP_MEMORY_SCOPE_AGENT);
}

// ---------------------------------------------------------------------------
// Kernel 6: out = bias (broadcast init before scatter).
// ---------------------------------------------------------------------------
__global__ void gat_bias_init(float* __restrict__ out, const float* __restrict__ bias, int total) {
    int i = blockIdx.x * blockDim.x + threadIdx.x;
    if (i < total) out[i] = bias[i & (IN_F - 1)];
}

// ---------------------------------------------------------------------------
// Kernel 7: scatter aggregation. One block (256 thr) per edge; thread c:
// out[dst, c] += feat[src, c] * (ex / denom[dst, h]),  h = c/64.
// ---------------------------------------------------------------------------
__global__ __launch_bounds__(256)
void gat_aggregate(const float* __restrict__ feat,
                   const int* __restrict__ src, const int* __restrict__ dst,
                   const float* __restrict__ ebuf, const float* __restrict__ denom,
                   float* __restrict__ out) {
    const int e = blockIdx.x;
    const int c = threadIdx.x;
    const int h = c >> 6;
    const int s = src[e], d = dst[e];
    const float alpha = ebuf[e * HEADS + h] / denom[d * HEADS + h];
    const float v = feat[(size_t)s * IN_F + c] * alpha;
    __hip_atomic_fetch_add(&out[(size_t)d * IN_F + c], v,
                           __ATOMIC_RELAXED, __HIP_MEMORY_SCOPE_AGENT);
}

// ---------------------------------------------------------------------------
extern "C" void kernel_launch(void* const* d_in, const int* in_sizes, int n_in,
                              void* d_out, int out_size, void* d_ws, size_t ws_size,
                              hipStream_t stream) {
    const float* emb    = (const float*)d_in[0];
    const int*   src    = (const int*)  d_in[1];
    const int*   dst    = (const int*)  d_in[2];
    const float* W      = (const float*)d_in[3];
    const float* attn_l = (const float*)d_in[4];
    const float* attn_r = (const float*)d_in[5];
    const float* bias   = (const float*)d_in[6];
    float* out = (float*)d_out;

    const int N = in_sizes[0] / IN_F;       // 50000
    const int E = in_sizes[1];              // 850000

    // Workspace carving (bytes): feat | el | er | m | denom | ebuf
    char* ws = (char*)d_ws;
    float* feat  = (float*)ws;                 ws += (size_t)N * IN_F * sizeof(float);
    float* el    = (float*)ws;                 ws += (size_t)N * HEADS * sizeof(float);
    float* er    = (float*)ws;                 ws += (size_t)N * HEADS * sizeof(float);
    float* m     = (float*)ws;                 ws += (size_t)N * HEADS * sizeof(float);
    float* denom = (float*)ws;                 ws += (size_t)N * HEADS * sizeof(float);
    float* ebuf  = (float*)ws;                 ws += (size_t)E * HEADS * sizeof(float);
    (void)ws_size; (void)n_in; (void)out_size;

    // 1. Projection GEMM (WMMA f32). N is a multiple of 16 (50000 = 3125*16).
    gat_gemm_wmma<<<N / 16, 128, 0, stream>>>(emb, W, feat);

    // 2. Per-node attention logits el/er.
    gat_elr<<<(N + 7) / 8, 256, 0, stream>>>(feat, attn_l, attn_r, el, er, N);

    // 3. Init segment-max / denom.
    const int NH = N * HEADS;
    gat_init_nh<<<(NH + 255) / 256, 256, 0, stream>>>(m, denom, NH);

    // 4. Edge logits + leaky relu + atomic fmax per (dst, head).
    const int E4 = E * HEADS;
    gat_logits<<<(E4 + 255) / 256, 256, 0, stream>>>(el, er, src, dst, ebuf, m, E4);

    // 5. exp + atomic denom accumulate.
    gat_expsum<<<(E4 + 255) / 256, 256, 0, stream>>>(dst, ebuf, m, denom, E4);

    // 6. out = bias.
    const int NO = N * IN_F;
    gat_bias_init<<<(NO + 255) / 256, 256, 0, stream>>>(out, bias, NO);

    // 7. Weighted scatter aggregation (one block per edge).
    gat_aggregate<<<E, 256, 0, stream>>>(feat, src, dst, ebuf, denom, out);
}